// TMKernelMF_34299608826382
// MI455X (gfx1250) — compile-verified
//
#include <hip/hip_runtime.h>
#include <math.h>

typedef __attribute__((ext_vector_type(2))) float v2f;
typedef __attribute__((ext_vector_type(8))) float v8f;

#define N_REP   96
#define KDIM    60
#define KPAD    64
#define XS_STR  68      // 68 % 64 == 4 -> conflict-free b64 lane loads
#define NN      96
#define GS_STR  97      // odd stride -> conflict-free column access
#define START   4096
#define NBATCH  12288
#define AUG_LOC 16384
#define AUG_C   61

// ---------------------------------------------------------------------------
// Prep: min over non-zero scales, per-column scaling vector, derived scalars.
// ws[0]=min_nz  ws[1]=1/ls^2  ws[2]=sigma_p[1]  ws[3]=sigma_p[3]  ws[4..63]=scaling
// ---------------------------------------------------------------------------
__global__ void __launch_bounds__(256) tm_prep_kernel(
    const float* __restrict__ kp,
    const float* __restrict__ scales,
    float* __restrict__ ws) {
  __shared__ float red[256];
  const int tid = threadIdx.x;
  float mn = INFINITY;
  for (int i = tid; i < NBATCH; i += 256) {
    float v = scales[START + i];
    if (v != 0.0f && v < mn) mn = v;
  }
  red[tid] = mn;
  __syncthreads();
  for (int s = 128; s > 0; s >>= 1) {
    if (tid < s) red[tid] = fminf(red[tid], red[tid + s]);
    __syncthreads();
  }
  if (tid == 0) {
    ws[0] = red[0];
    // ls = exp(lengthscale[r=1]) * sqrt(2*SMOOTH) = exp(kp[11]) * sqrt(3)
    float ls = expf(kp[11]) * sqrtf(3.0f);
    ws[1] = 1.0f / (ls * ls);
    ws[2] = kp[1];   // sigma_params[r]
    ws[3] = kp[3];   // sigma_params[r+R]
  }
  if (tid < 30) {
    float kk = (float)(tid + 1);
    ws[4 + tid] = expf(kp[5] - 0.5f * expf(kp[7]) * kk);        // scaling_now
  } else if (tid < 60) {
    float kk = (float)(tid - 29);
    ws[4 + tid] = expf(kp[8] - 0.5f * expf(kp[9]) * kk);        // scaling_preb
  }
}

// ---------------------------------------------------------------------------
// One workgroup (8 wave32) per batch location:
//   load+scale X into LDS -> WMMA f32 gram -> elementwise kernel transform
//   -> in-LDS Cholesky -> stream g and chol to HBM.
// ---------------------------------------------------------------------------
__global__ void __launch_bounds__(256) tm_batch_kernel(
    const float* __restrict__ aug,
    const float* __restrict__ scales,
    const float* __restrict__ nug,
    const float* __restrict__ ws,
    float* __restrict__ out) {
  __shared__ __align__(16) float Xs[N_REP * XS_STR];   // 26112 B
  __shared__ __align__(16) float Gs[NN * GS_STR];      // 37248 B
  __shared__ float sq[NN];
  __shared__ float s_scal[4];                          // inv_ls2, sigma2, inv_nug, diag_inv
  __shared__ float s_scaling[KPAD];

  const int b   = blockIdx.x;
  const int tid = threadIdx.x;

  if (tid < KPAD) s_scaling[tid] = (tid < KDIM) ? ws[4 + tid] : 0.0f;
  if (tid == 0) {
    float sc = scales[START + b];
    if (sc == 0.0f) sc = ws[0] * 0.5f;
    float sg = expf(ws[2] + ws[3] * logf(sc));
    s_scal[0] = ws[1];
    s_scal[1] = sg * sg;
    s_scal[2] = 1.0f / nug[b];
    out[2 * (size_t)NBATCH * NN * NN + b] = nug[b];    // nug_means passthrough
  }
  __syncthreads();

  // ---- load X (96 x 64, cols 60..63 zero); aug cols 1..60 are contiguous ----
  const size_t loc_base = (size_t)(START + b) * AUG_C;
  for (int idx = tid; idx < N_REP * KPAD; idx += 256) {
    const int rep = idx >> 6;
    const int k   = idx & 63;
    float v = 0.0f;
    if (k < KDIM) {
      float raw = aug[(size_t)rep * (AUG_LOC * AUG_C) + loc_base + 1 + k];
      v = isnan(raw) ? 0.0f : raw;
      v *= s_scaling[k];
    }
    Xs[rep * XS_STR + k] = v;
  }
  __syncthreads();

  // ---- gram = X * X^T via V_WMMA_F32_16X16X4_F32, 6x6 tiles of 16x16 ----
  const int wave = tid >> 5;
  const int lane = tid & 31;
  const int half = lane >> 4;
  const int m    = lane & 15;
  for (int t = wave; t < 36; t += 8) {
    const int i0 = (t / 6) * 16;
    const int j0 = (t % 6) * 16;
    v8f acc = {};
    const float* pa = &Xs[(i0 + m) * XS_STR + 2 * half];
    const float* pb = &Xs[(j0 + m) * XS_STR + 2 * half];
    for (int k = 0; k < KPAD; k += 4) {
      v2f a  = *(const v2f*)(pa + k);
      v2f bb = *(const v2f*)(pb + k);
      acc = __builtin_amdgcn_wmma_f32_16x16x4_f32(
          /*neg_a=*/false, a, /*neg_b=*/false, bb,
          /*c_mod=*/(short)0, acc, /*reuse_a=*/false, /*reuse_b=*/false);
    }
#pragma unroll
    for (int v = 0; v < 8; ++v)
      Gs[(i0 + v + 8 * half) * GS_STR + (j0 + m)] = acc[v];
  }
  __syncthreads();

  // ---- sq_i = diag(gram)_i / ls^2 ----
  if (tid < NN) sq[tid] = Gs[tid * GS_STR + tid] * s_scal[0];
  __syncthreads();

  // ---- elementwise transform + g output ----
  const float inv_ls2 = s_scal[0];
  const float sigma2  = s_scal[1];
  const float inv_nug = s_scal[2];
  float* outg = out + (size_t)b * (NN * NN);
  for (int idx = tid; idx < NN * NN; idx += 256) {
    const int i = idx / NN;
    const int j = idx - i * NN;
    float lin = Gs[i * GS_STR + j];
    float d2  = sq[i] + sq[j] - 2.0f * lin * inv_ls2;
    d2 = fmaxf(d2, 0.0f);
    float c  = sqrtf(3.0f * d2);
    float nl = (1.0f + c) * expf(-c);
    float g  = (lin + sigma2 * nl) * inv_nug + ((i == j) ? 1.0f : 0.0f);
    if (b == 0) g = (i == j) ? 1.0f : 0.0f;   // g[0] = I
    Gs[i * GS_STR + j] = g;
    outg[idx] = g;
  }
  __syncthreads();

  // ---- in-LDS right-looking Cholesky (lower) ----
  for (int k = 0; k < NN; ++k) {
    if (tid == 0) {
      float dk = sqrtf(Gs[k * GS_STR + k]);
      Gs[k * GS_STR + k] = dk;
      s_scal[3] = 1.0f / dk;
    }
    __syncthreads();
    const float dinv = s_scal[3];
    for (int i = k + 1 + tid; i < NN; i += 256)
      Gs[i * GS_STR + k] *= dinv;
    __syncthreads();
    const int mrem  = NN - 1 - k;
    const int total = mrem * (mrem + 1) / 2;
    for (int idx = tid; idx < total; idx += 256) {
      // triangular index -> (row r, col jj) in trailing block
      float tf = sqrtf(8.0f * (float)idx + 1.0f);
      int r = (int)((tf - 1.0f) * 0.5f);
      while ((r + 1) * (r + 2) / 2 <= idx) ++r;
      while (r * (r + 1) / 2 > idx) --r;
      const int jj = idx - r * (r + 1) / 2;
      const int i = k + 1 + r;
      const int j = k + 1 + jj;
      Gs[i * GS_STR + j] -= Gs[i * GS_STR + k] * Gs[j * GS_STR + k];
    }
    __syncthreads();
  }

  // ---- chol output (zero upper triangle) ----
  float* outc = out + (size_t)NBATCH * (NN * NN) + (size_t)b * (NN * NN);
  for (int idx = tid; idx < NN * NN; idx += 256) {
    const int i = idx / NN;
    const int j = idx - i * NN;
    outc[idx] = (j <= i) ? Gs[i * GS_STR + j] : 0.0f;
  }
}

extern "C" void kernel_launch(void* const* d_in, const int* in_sizes, int n_in,
                              void* d_out, int out_size, void* d_ws, size_t ws_size,
                              hipStream_t stream) {
  const float* kp     = (const float*)d_in[0];  // kernel_params (12)
  const float* aug    = (const float*)d_in[1];  // augmented_response (96,16384,61)
  const float* scales = (const float*)d_in[2];  // scales (16384)
  const float* nug    = (const float*)d_in[3];  // nug_means (12288)
  float* ws  = (float*)d_ws;
  float* out = (float*)d_out;

  tm_prep_kernel<<<1, 256, 0, stream>>>(kp, scales, ws);
  tm_batch_kernel<<<NBATCH, 256, 0, stream>>>(aug, scales, nug, ws, out);
}